// FastGTN_86311662780807
// MI455X (gfx1250) — compile-verified
//
#include <hip/hip_runtime.h>

#define RR 5       // relations
#define CC 2       // channels
#define LL 2       // layers
#define HIDD 128   // hidden dim

typedef __attribute__((ext_vector_type(2))) float v2f;
typedef __attribute__((ext_vector_type(8))) float v8f;

// ---------------------------------------------------------------------------
// Tiny softmax over the (L, C, R) relation-weight tensor -> filt[L*C*R]
// ---------------------------------------------------------------------------
__global__ void softmax_filt_kernel(const float* __restrict__ gt_w,
                                    float* __restrict__ filt) {
  int t = threadIdx.x;           // t = l*CC + c
  if (t >= LL * CC) return;
  const float* g = gt_w + t * RR;
  float m = g[0];
  for (int r = 1; r < RR; ++r) m = fmaxf(m, g[r]);
  float e[RR];
  float s = 0.f;
  for (int r = 0; r < RR; ++r) { e[r] = __expf(g[r] - m); s += e[r]; }
  float inv = 1.0f / s;
  for (int r = 0; r < RR; ++r) filt[t * RR + r] = e[r] * inv;
}

// ---------------------------------------------------------------------------
// Per-destination degree: deg[c][n] = sum over incoming edges of filt weight
// ---------------------------------------------------------------------------
__global__ void deg_kernel(const int* __restrict__ dst,
                           const int* __restrict__ etype,
                           const float* __restrict__ filt,
                           float* __restrict__ deg,
                           int E, int N, int layer) {
  int e = blockIdx.x * blockDim.x + threadIdx.x;
  if (e >= E) return;
  int d  = dst[e];
  int et = etype[e];
  #pragma unroll
  for (int c = 0; c < CC; ++c)
    atomicAdd(&deg[(size_t)c * N + d], filt[(layer * CC + c) * RR + et]);
}

// ---------------------------------------------------------------------------
// Message passing: one wave (32 lanes) per edge, 4 floats of H per lane.
// Hout[c][d][:] += Hin[c][s][:] * (w / deg[c][d])
// ---------------------------------------------------------------------------
__global__ void msg_kernel(const int* __restrict__ src,
                           const int* __restrict__ dst,
                           const int* __restrict__ etype,
                           const float* __restrict__ filt,
                           const float* __restrict__ deg,
                           const float* __restrict__ Hin,
                           float* __restrict__ Hout,
                           int E, int N, int layer) {
  int gid  = blockIdx.x * blockDim.x + threadIdx.x;
  int e    = gid >> 5;
  int lane = gid & 31;
  if (e >= E) return;
  int s  = src[e];
  int d  = dst[e];
  int et = etype[e];
  #pragma unroll
  for (int c = 0; c < CC; ++c) {
    float w  = filt[(layer * CC + c) * RR + et];
    float wn = w / deg[(size_t)c * N + d];
    const float4* hp = (const float4*)(Hin + ((size_t)c * N + s) * HIDD);
    float4 v = hp[lane];
    float* op = Hout + ((size_t)c * N + d) * HIDD + lane * 4;
    atomicAdd(op + 0, v.x * wn);
    atomicAdd(op + 1, v.y * wn);
    atomicAdd(op + 2, v.z * wn);
    atomicAdd(op + 3, v.w * wn);
  }
}

// ---------------------------------------------------------------------------
// Generic fp32 WMMA GEMM:  Out[batch] = act(A' @ B[batch] + bias)
//   A'[row][k] = A[row*aRowStride + (k>>aKShift)*aChunkStride + (k&aKMask)]
// (folds transpose(H,(1,0,2)).reshape into the A read).
//
// 256 threads = 8 waves; each wave: one 16-row M tile, all NTILES col tiles.
// B staged in LDS in 32-deep K panels, laid out as (k-pair, n) float2 so each
// WMMA B fragment is a single aligned ds_load_b64 at an immediate offset.
// A panel base is hoisted (chunk index constant within a 32-K panel since
// both chunk sizes are multiples of 32), so A reads are clauseable b64 loads
// at immediate offsets.
//
// V_WMMA_F32_16X16X4_F32 layouts (ISA 7.12.2):
//   A (16x4): VGPR0 = K{0 | 2}, VGPR1 = K{1 | 3} for lanes {0-15 | 16-31}
//   B (4x16): VGPR0 = K{0 | 2}, VGPR1 = K{1 | 3}, N = lane&15
//   C/D     : VGPRv = rows {v | v+8}, N = lane&15
// ---------------------------------------------------------------------------
template <int NTILES>
__global__ __launch_bounds__(256) void gemm_wmma_f32(
    const float* __restrict__ A, const float* __restrict__ B,
    const float* __restrict__ bias, float* __restrict__ Out,
    int M, int K,
    int aRowStride, int aKShift, int aKMask, size_t aChunkStride,
    size_t bBatchStride, size_t outBatchStride, int relu) {
  constexpr int NDIM = NTILES * 16;
  // (k-pair, n) float2 layout: pair kp holds B[2*kp][n], B[2*kp+1][n]
  __shared__ float2 Bs[16 * NDIM];

  const int wave = threadIdx.x >> 5;
  const int lane = threadIdx.x & 31;
  const int half = lane & 15;   // M row (A), N col (B/C/D) within tile
  const int hi   = lane >> 4;   // half-wave select

  const float* Bb = B + bBatchStride * blockIdx.y;
  float*       Ob = Out + outBatchStride * blockIdx.y;

  const int mTile = blockIdx.x * 8 + wave;
  const int row0  = mTile * 16;

  v8f acc[NTILES];
  for (int nt = 0; nt < NTILES; ++nt) {
    float bv = bias ? bias[nt * 16 + half] : 0.0f;
    v8f t;
    #pragma unroll
    for (int i = 0; i < 8; ++i) t[i] = bv;
    acc[nt] = t;
  }

  int aRow = row0 + half;
  if (aRow >= M) aRow = M - 1;                 // clamp: keeps barriers uniform
  const float* Arow = A + (size_t)aRow * aRowStride;

  // per-lane LDS fragment base: row-pair skew for the upper half-wave
  const float2* BsLane = Bs + (size_t)hi * NDIM + half;

  for (int k0 = 0; k0 < K; k0 += 32) {
    // cooperative B panel load -> LDS (float2 k-pairs, contiguous b64 stores)
    for (int t = threadIdx.x; t < 16 * NDIM; t += 256) {
      int kp = t / NDIM, nn = t % NDIM;
      float2 v;
      v.x = Bb[(size_t)(k0 + 2 * kp)     * NDIM + nn];
      v.y = Bb[(size_t)(k0 + 2 * kp + 1) * NDIM + nn];
      Bs[(size_t)kp * NDIM + nn] = v;
    }
    __syncthreads();

    // hoisted A panel base (chunk index constant across the 32-K panel)
    const float* ap0 = Arow
        + (size_t)(k0 >> aKShift) * aChunkStride
        + ((k0 & aKMask) + 2 * hi);

    #pragma unroll
    for (int kk = 0; kk < 32; kk += 4) {
      v2f a = *(const v2f*)(ap0 + kk);         // aligned b64, immediate offset
      #pragma unroll
      for (int nt = 0; nt < NTILES; ++nt) {
        v2f b = *(const v2f*)(BsLane + (size_t)(kk >> 1) * NDIM + nt * 16);
        acc[nt] = __builtin_amdgcn_wmma_f32_16x16x4_f32(
            false, a, false, b, (short)0, acc[nt], false, false);
      }
    }
    __syncthreads();
  }

  // epilogue: optional relu, guarded store
  for (int nt = 0; nt < NTILES; ++nt) {
    #pragma unroll
    for (int vr = 0; vr < 8; ++vr) {
      int r = row0 + vr + 8 * hi;
      if (r < M) {
        float v = acc[nt][vr];
        if (relu) v = fmaxf(v, 0.0f);
        Ob[(size_t)r * NDIM + nt * 16 + half] = v;
      }
    }
  }
}

// ---------------------------------------------------------------------------
// Final gather: out[i][j] = Y[category_idx[i]][j]
// ---------------------------------------------------------------------------
__global__ void gather_kernel(const float* __restrict__ Y,
                              const int* __restrict__ cat,
                              float* __restrict__ out, int ncat) {
  int t = blockIdx.x * blockDim.x + threadIdx.x;
  if (t >= ncat * 16) return;
  int i = t >> 4, j = t & 15;
  out[t] = Y[(size_t)cat[i] * 16 + j];
}

// ---------------------------------------------------------------------------
extern "C" void kernel_launch(void* const* d_in, const int* in_sizes, int n_in,
                              void* d_out, int out_size, void* d_ws, size_t ws_size,
                              hipStream_t stream) {
  const float* h   = (const float*)d_in[0];
  const int*   src = (const int*)d_in[1];
  const int*   dst = (const int*)d_in[2];
  const int*   et  = (const int*)d_in[3];
  const int*   cat = (const int*)d_in[4];
  const float* gtw = (const float*)d_in[5];
  const float* Wc  = (const float*)d_in[6];
  const float* W1  = (const float*)d_in[7];
  const float* b1  = (const float*)d_in[8];
  const float* W2  = (const float*)d_in[9];
  const float* b2  = (const float*)d_in[10];

  const int IN_DIM = 256;
  const int N    = in_sizes[0] / IN_DIM;
  const int E    = in_sizes[1];
  const int NCAT = in_sizes[4];

  // workspace layout (floats)
  float* ws    = (float*)d_ws;
  float* filt  = ws;                               // 32 (L*C*R = 20 used)
  float* deg   = filt + 32;                        // CC*N
  float* Hbuf0 = deg + (size_t)CC * N;             // CC*N*HIDD
  float* Hbuf1 = Hbuf0 + (size_t)CC * N * HIDD;    // CC*N*HIDD
  float* Z     = Hbuf1 + (size_t)CC * N * HIDD;    // N*HIDD
  float* Y     = Z + (size_t)N * HIDD;             // N*16

  // 1) relation softmax (tiny)
  softmax_filt_kernel<<<1, 32, 0, stream>>>(gtw, filt);

  const int mtiles = (N + 15) / 16;     // 3125 (N = 50000 is 16-aligned)
  const int gx     = (mtiles + 7) / 8;  // 8 waves per block

  // 2) H[c] = h @ Wc[c]  (batched over channel via blockIdx.y)
  gemm_wmma_f32<8><<<dim3(gx, CC), 256, 0, stream>>>(
      h, Wc, nullptr, Hbuf0, N, IN_DIM,
      /*aRowStride*/ IN_DIM, /*aKShift*/ 8, /*aKMask*/ 255, /*aChunkStride*/ 0,
      /*bBatch*/ (size_t)IN_DIM * HIDD, /*outBatch*/ (size_t)N * HIDD, /*relu*/ 0);

  // 3) two rounds of normalized relation-weighted propagation
  float* Hcur = Hbuf0;
  float* Hnxt = Hbuf1;
  for (int l = 0; l < LL; ++l) {
    hipMemsetAsync(deg, 0, sizeof(float) * (size_t)CC * N, stream);
    hipMemsetAsync(Hnxt, 0, sizeof(float) * (size_t)CC * N * HIDD, stream);
    deg_kernel<<<(E + 255) / 256, 256, 0, stream>>>(dst, et, filt, deg, E, N, l);
    long long th = (long long)E * 32;  // one wave per edge
    msg_kernel<<<(unsigned)((th + 255) / 256), 256, 0, stream>>>(
        src, dst, et, filt, deg, Hcur, Hnxt, E, N, l);
    float* tmp = Hcur; Hcur = Hnxt; Hnxt = tmp;
  }

  // 4) Z = relu(X @ W1 + b1), X[n][c*128+hh] = Hcur[c][n][hh] folded into A-read
  gemm_wmma_f32<8><<<dim3(gx, 1), 256, 0, stream>>>(
      Hcur, W1, b1, Z, N, CC * HIDD,
      /*aRowStride*/ HIDD, /*aKShift*/ 7, /*aKMask*/ 127,
      /*aChunkStride*/ (size_t)N * HIDD,
      0, 0, /*relu*/ 1);

  // 5) Y = Z @ W2 + b2  (single 16-wide N tile)
  gemm_wmma_f32<1><<<dim3(gx, 1), 256, 0, stream>>>(
      Z, W2, b2, Y, N, HIDD,
      /*aRowStride*/ HIDD, /*aKShift*/ 7, /*aKMask*/ 127, /*aChunkStride*/ 0,
      0, 0, /*relu*/ 0);

  // 6) gather rows for category_idx
  gather_kernel<<<(NCAT * 16 + 255) / 256, 256, 0, stream>>>(
      Y, cat, (float*)d_out, NCAT);
}